// RAN_7868380086984
// MI455X (gfx1250) — compile-verified
//
#include <hip/hip_runtime.h>

typedef __attribute__((ext_vector_type(2))) float v2f;
typedef __attribute__((ext_vector_type(4))) float v4f;
typedef __attribute__((ext_vector_type(8))) float v8f;

#define HDIM 256
#define NB   16
#define HW   65536  // 256*256

// ---------------------------------------------------------------------------
// One MLP layer tile pass: Y[:, tiles-of-this-wave] = act(X @ W + bias)
// X is 16 x K in LDS (row stride ldx). W is Kvalid x 256 in global memory
// (rows >= Kvalid treated as zero). Each of the 8 waves computes two 16x16
// output tiles using V_WMMA_F32_16X16X4_F32 (fp32 in, fp32 accumulate).
// ---------------------------------------------------------------------------
__device__ __forceinline__ void mlp_layer(const float* __restrict__ Xl, int ldx,
                                          int K, int Kvalid,
                                          const float* __restrict__ W,
                                          const float* __restrict__ bias,
                                          float* __restrict__ Yl,   // LDS out (or null)
                                          float* __restrict__ Yg,   // global out (or null)
                                          bool relu, int lane, int wave)
{
  const int m16  = lane & 15;          // row (A) / col-in-tile (B, C/D)
  const int koff = (lane >> 4) << 1;   // lanes 16-31 hold K+2, K+3

  for (int t = 0; t < 2; ++t) {
    const int n = (((t << 3) + wave) << 4) + m16;   // global output column
    v8f acc = {};
    for (int k0 = 0; k0 < K; k0 += 4) {
      const int ka = k0 + koff;
      v2f A, B;
      // A: 16x4 fragment from LDS activations
      A.x = Xl[m16 * ldx + ka];
      A.y = Xl[m16 * ldx + ka + 1];
      // B: 4x16 fragment from global weights (guard padded K rows)
      B.x = (ka     < Kvalid) ? W[(size_t)ka       * HDIM + n] : 0.0f;
      B.y = (ka + 1 < Kvalid) ? W[(size_t)(ka + 1) * HDIM + n] : 0.0f;
      acc = __builtin_amdgcn_wmma_f32_16x16x4_f32(
          /*neg_a=*/false, A, /*neg_b=*/false, B,
          /*c_mod=*/(short)0, acc, /*reuse_a=*/false, /*reuse_b=*/false);
    }
    const float bv = bias[n];
#pragma unroll
    for (int v = 0; v < 8; ++v) {
      const int mrow = v + ((lane >> 4) << 3);   // C/D layout: VGPR v -> M=v (+8 hi half)
      float r = acc[v] + bv;
      if (relu) r = fmaxf(r, 0.0f);
      if (Yl) Yl[mrow * HDIM + n] = r;
      if (Yg) Yg[mrow * HDIM + n] = r;
    }
  }
}

// ---------------------------------------------------------------------------
// Kernel 1: whole conditioning MLP in one workgroup (8 waves, wave32).
// Produces g (gamma) and b (beta), each 16 x 256, into workspace.
// ---------------------------------------------------------------------------
__global__ __launch_bounds__(256) void spade_mlp_kernel(
    const float* __restrict__ seg,
    const float* __restrict__ tw0, const float* __restrict__ tb0,
    const float* __restrict__ tw1, const float* __restrict__ tb1,
    const float* __restrict__ tw2, const float* __restrict__ tb2,
    const float* __restrict__ tw3, const float* __restrict__ tb3,
    const float* __restrict__ gw0, const float* __restrict__ gb0,
    const float* __restrict__ gw1, const float* __restrict__ gb1,
    const float* __restrict__ gw2, const float* __restrict__ gb2,
    const float* __restrict__ gw3, const float* __restrict__ gb3,
    const float* __restrict__ bw0, const float* __restrict__ bb0,
    const float* __restrict__ bw1, const float* __restrict__ bb1,
    const float* __restrict__ bw2, const float* __restrict__ bb2,
    const float* __restrict__ bw3, const float* __restrict__ bb3,
    float* __restrict__ g_out, float* __restrict__ b_out)
{
  __shared__ float segl[NB * 12];        // segmap zero-padded K=11 -> 12
  __shared__ float bufA[NB * HDIM];
  __shared__ float bufB[NB * HDIM];
  __shared__ float bufH[NB * HDIM];      // shared trunk output h

  const int tid  = threadIdx.x;
  const int lane = tid & 31;
  const int wave = tid >> 5;

  if (tid < NB * 12) {
    const int m = tid / 12, k = tid % 12;
    segl[tid] = (k < 11) ? seg[m * 11 + k] : 0.0f;
  }
  __syncthreads();

  // trunk: 4 layers with ReLU
  mlp_layer(segl, 12, 12, 11, tw0, tb0, bufA, nullptr, true, lane, wave); __syncthreads();
  mlp_layer(bufA, HDIM, HDIM, HDIM, tw1, tb1, bufB, nullptr, true, lane, wave); __syncthreads();
  mlp_layer(bufB, HDIM, HDIM, HDIM, tw2, tb2, bufA, nullptr, true, lane, wave); __syncthreads();
  mlp_layer(bufA, HDIM, HDIM, HDIM, tw3, tb3, bufH, nullptr, true, lane, wave); __syncthreads();

  // gamma branch
  mlp_layer(bufH, HDIM, HDIM, HDIM, gw0, gb0, bufA, nullptr, true, lane, wave); __syncthreads();
  mlp_layer(bufA, HDIM, HDIM, HDIM, gw1, gb1, bufB, nullptr, true, lane, wave); __syncthreads();
  mlp_layer(bufB, HDIM, HDIM, HDIM, gw2, gb2, bufA, nullptr, true, lane, wave); __syncthreads();
  mlp_layer(bufA, HDIM, HDIM, HDIM, gw3, gb3, nullptr, g_out, false, lane, wave); __syncthreads();

  // beta branch
  mlp_layer(bufH, HDIM, HDIM, HDIM, bw0, bb0, bufA, nullptr, true, lane, wave); __syncthreads();
  mlp_layer(bufA, HDIM, HDIM, HDIM, bw1, bb1, bufB, nullptr, true, lane, wave); __syncthreads();
  mlp_layer(bufB, HDIM, HDIM, HDIM, bw2, bb2, bufA, nullptr, true, lane, wave); __syncthreads();
  mlp_layer(bufA, HDIM, HDIM, HDIM, bw3, bb3, nullptr, b_out, false, lane, wave);
}

// ---------------------------------------------------------------------------
// Kernel 2: instance norm + spatially-adaptive affine.
// One block per (n,c) plane (1024 blocks). 256 threads * 64 float4 each.
// Pass 1 computes mean/var; pass 2 re-reads (L2-resident) and writes with
// non-temporal stores so the output stream doesn't evict pass-2 inputs.
// ---------------------------------------------------------------------------
__global__ __launch_bounds__(256) void spade_norm_kernel(
    const float* __restrict__ x,
    const float* __restrict__ g,
    const float* __restrict__ b,
    float* __restrict__ out)
{
  const int bid = blockIdx.x;          // (n, c) plane index
  const int n   = bid >> 6;            // / C
  const size_t base4 = (size_t)bid * (HW / 4);
  const v4f* __restrict__ xp = (const v4f*)x + base4;
  v4f* __restrict__ op       = (v4f*)out + base4;

  const int tid = threadIdx.x;

  // ---- pass 1: sum / sum-of-squares over the 65536-element plane ----
  float sum = 0.0f, ssq = 0.0f;
#pragma unroll 8
  for (int i = 0; i < 64; ++i) {
    const v4f v = xp[tid + (i << 8)];
    sum += v.x + v.y + v.z + v.w;
    ssq += v.x * v.x + v.y * v.y + v.z * v.z + v.w * v.w;
  }
  // wave32 reduction
#pragma unroll
  for (int off = 16; off > 0; off >>= 1) {
    sum += __shfl_xor(sum, off, 32);
    ssq += __shfl_xor(ssq, off, 32);
  }
  __shared__ float wsum[8], wssq[8];
  __shared__ float stats[2];
  if ((tid & 31) == 0) { wsum[tid >> 5] = sum; wssq[tid >> 5] = ssq; }
  __syncthreads();
  if (tid == 0) {
    float s = 0.0f, q = 0.0f;
#pragma unroll
    for (int i = 0; i < 8; ++i) { s += wsum[i]; q += wssq[i]; }
    const float mean = s * (1.0f / (float)HW);
    const float var  = q * (1.0f / (float)HW) - mean * mean;
    stats[0] = mean;
    stats[1] = rsqrtf(var + 1e-5f);
  }
  __syncthreads();
  const float mean = stats[0];
  const float rstd = stats[1];

  // ---- per-row (H) scale/shift: out = x*s[h] + t[h] ----
  __shared__ float sh_s[HDIM], sh_t[HDIM];
  {
    const int h = tid;
    const float s = rstd * (1.0f + g[n * HDIM + h]);
    sh_s[h] = s;
    sh_t[h] = b[n * HDIM + h] - mean * s;
  }
  __syncthreads();

  // ---- pass 2: normalize + affine, streaming stores ----
#pragma unroll 8
  for (int i = 0; i < 64; ++i) {
    const int j = tid + (i << 8);
    const int h = j >> 6;                       // 64 float4 per row
    const v4f v = __builtin_nontemporal_load(&xp[j]);  // last use of x
    const float s = sh_s[h], t = sh_t[h];
    v4f r;
    r.x = fmaf(v.x, s, t);
    r.y = fmaf(v.y, s, t);
    r.z = fmaf(v.z, s, t);
    r.w = fmaf(v.w, s, t);
    __builtin_nontemporal_store(r, &op[j]);     // never re-read
  }
}

// ---------------------------------------------------------------------------
extern "C" void kernel_launch(void* const* d_in, const int* in_sizes, int n_in,
                              void* d_out, int out_size, void* d_ws, size_t ws_size,
                              hipStream_t stream) {
  (void)in_sizes; (void)n_in; (void)out_size; (void)ws_size;

  const float* x   = (const float*)d_in[0];
  const float* seg = (const float*)d_in[1];
  const float* p[24];
  for (int i = 0; i < 24; ++i) p[i] = (const float*)d_in[2 + i];

  float* g_out = (float*)d_ws;                  // 16*256 floats
  float* b_out = g_out + NB * HDIM;             // 16*256 floats

  spade_mlp_kernel<<<1, 256, 0, stream>>>(
      seg,
      p[0],  p[1],  p[2],  p[3],  p[4],  p[5],  p[6],  p[7],
      p[8],  p[9],  p[10], p[11], p[12], p[13], p[14], p[15],
      p[16], p[17], p[18], p[19], p[20], p[21], p[22], p[23],
      g_out, b_out);

  spade_norm_kernel<<<NB * 64, 256, 0, stream>>>(x, g_out, b_out, (float*)d_out);
}